// GaussianSplatting2D_8022998909463
// MI455X (gfx1250) — compile-verified
//
#include <hip/hip_runtime.h>

// ---------------------------------------------------------------------------
// 2D Gaussian splatting, CDNA5 (gfx1250, wave32).
// Exponent matrix [P,N] computed as [P,8]x[8,N] matmul via V_WMMA_F32_16X16X4_F32.
// Top-k runs in the log2 domain (exp2 is monotonic) -> only 8 v_exp_f32 per pixel.
// A-operand LDS loads are software-pipelined one chunk ahead to hide ds latency.
// ---------------------------------------------------------------------------

typedef __attribute__((ext_vector_type(2))) float v2f;
typedef __attribute__((ext_vector_type(8))) float v8f;

#define NG     2048
#define IMG_W  256
#define HW     65536          // 256*256
#define LOG2E  1.44269504088896340736f
#define NEG_BIG (-3.0e38f)

__device__ __forceinline__ float swz16_f(float v) { return __shfl_xor(v, 16, 32); }
__device__ __forceinline__ int   swz16_i(int   v) { return __shfl_xor(v, 16, 32); }

__launch_bounds__(256, 2)
__global__ void gsplat_wmma_kernel(const float* __restrict__ xy,
                                   const float* __restrict__ scale,
                                   const float* __restrict__ rot,
                                   const float* __restrict__ feat,
                                   float* __restrict__ out)
{
    // Folded quadratic-form coefficients, 8 floats per Gaussian (K padded to 8).
    __shared__ float s_coef[NG * 8];   // 64 KB of the 320 KB WGP LDS

    // ---- build coefficients in LDS (cheap: 8 sincos per thread) ----
    for (int g = (int)threadIdx.x; g < NG; g += 256) {
        float gx = xy[2*g + 0], gy = xy[2*g + 1];
        float sx = scale[2*g + 0], sy = scale[2*g + 1];
        float r  = rot[g];
        float c  = __cosf(r), s = __sinf(r);
        float sx2 = sx*sx, sy2 = sy*sy, c2 = c*c, s2 = s*s;
        float A  = sx2*c2 + sy2*s2;            // dx^2 coefficient
        float B  = sx2*s2 + sy2*c2;            // dy^2 coefficient
        float Cc = 2.0f*c*s*(sx2 - sy2);       // dx*dy coefficient
        const float k = -0.5f * LOG2E;         // fold exp(-0.5*Q) -> exp2(k*Q')
        float* cf = &s_coef[g * 8];
        cf[0] = k * A;                                     // * px*px
        cf[1] = k * B;                                     // * py*py
        cf[2] = k * Cc;                                    // * px*py
        cf[3] = k * (-2.0f*A*gx - Cc*gy);                  // * px
        cf[4] = k * (-2.0f*B*gy - Cc*gx);                  // * py
        cf[5] = k * (A*gx*gx + B*gy*gy + Cc*gx*gy);        // * 1
        cf[6] = 0.0f;                                      // K pad
        cf[7] = 0.0f;                                      // K pad
    }
    __syncthreads();

    const int lane = threadIdx.x & 31;
    const int wv   = threadIdx.x >> 5;   // wave in block: 0..7
    const int hif  = lane >> 4;          // lane half (K/row split)
    const int col  = lane & 15;          // pixel column / gaussian row within tile

    // 16 consecutive pixels per wave, 128 per block.
    const int   p  = blockIdx.x * 128 + wv * 16 + col;
    const float px = ((p & (IMG_W - 1)) + 0.5f) * (1.0f / 256.0f);
    const float py = ((p >> 8)          + 0.5f) * (1.0f / 256.0f);

    // B operand (monomials, K x 16 pixels), fixed for the whole kernel.
    // 32-bit B layout: lanes 0-15 hold K0 (vgpr0) / K1 (vgpr1); lanes 16-31 K2/K3.
    v2f b0, b1;
    b0.x = hif ? px*py : px*px;   // K2 : K0
    b0.y = hif ? px    : py*py;   // K3 : K1
    b1.x = hif ? 0.0f  : py;      // K6 : K4
    b1.y = hif ? 0.0f  : 1.0f;    // K7 : K5

    // Per-lane sorted (descending) top-8 of this lane's 1024 log2-weights.
    float tv[8]; int ti[8];
#pragma unroll
    for (int t = 0; t < 8; ++t) { tv[t] = NEG_BIG; ti[t] = 0; }

    // Prime the software pipeline: A-operand slices for chunk 0.
    const int abase = col * 8 + hif * 2;       // within a 16-gaussian chunk
    v2f a0 = *(const v2f*)(&s_coef[abase]);
    v2f a1 = *(const v2f*)(&s_coef[abase + 4]);

#pragma unroll 1
    for (int ch = 0; ch < NG / 16; ++ch) {
        // Prefetch next chunk's A operand; its s_wait_dscnt lands after this
        // chunk's top-k work, hiding LDS latency. ((ch+1)&127 keeps it in range.)
        const int nbase = ((ch + 1) & 127) * 128 + abase;
        v2f na0 = *(const v2f*)(&s_coef[nbase]);
        v2f na1 = *(const v2f*)(&s_coef[nbase + 4]);

        v8f acc = {};
        acc = __builtin_amdgcn_wmma_f32_16x16x4_f32(false, a0, false, b0,
                                                    (short)0, acc, false, false);
        acc = __builtin_amdgcn_wmma_f32_16x16x4_f32(false, a1, false, b1,
                                                    (short)0, acc, false, false);

        // acc[j] = log2-weight of gaussian row (hif*8 + j) for pixel col.
        // exp2 is monotonic -> rank directly on acc, defer exp2 to the end.
        float amax = fmaxf(acc[0], acc[1]);
        amax = fmaxf(amax, fmaxf(acc[2], acc[3]));
        amax = fmaxf(amax, fmaxf(acc[4], acc[5]));
        amax = fmaxf(amax, fmaxf(acc[6], acc[7]));

        if (amax > tv[7]) {                      // chunk can contribute
            const int gb = ch * 16 + hif * 8;
#pragma unroll
            for (int j = 0; j < 8; ++j) {
                float cv = acc[j];
                if (cv > tv[7]) {                // branchless sorted insert
                    int ci = gb + j;
#pragma unroll
                    for (int t = 0; t < 8; ++t) {
                        bool  sw = cv > tv[t];
                        float ov = tv[t]; int oi = ti[t];
                        tv[t] = sw ? cv : ov;  ti[t] = sw ? ci : oi;
                        cv    = sw ? ov : cv;  ci    = sw ? oi : ci;
                    }
                }
            }
        }

        a0 = na0;  a1 = na1;
    }

    // Merge the two lane halves of each pixel (lane n rows 0..7, lane n+16 rows 8..15).
    // Both lists sorted descending -> top8(A u B) = { max(A[t], B[7-t]) }.
    float mv[8]; int mi[8];
#pragma unroll
    for (int t = 0; t < 8; ++t) {
        float ov = swz16_f(tv[7 - t]);
        int   oi = swz16_i(ti[7 - t]);
        bool keep = tv[t] >= ov;
        mv[t] = keep ? tv[t] : ov;
        mi[t] = keep ? ti[t] : oi;
    }

    // Only now leave the log2 domain: 8 exp2 per pixel instead of 2048.
    float w[8];
    float sum = 0.0f;
#pragma unroll
    for (int t = 0; t < 8; ++t) {
        w[t] = __builtin_amdgcn_exp2f(mv[t]);
        sum += w[t];
    }
    const float inv = 1.0f / (sum + 1e-4f);

    float o0 = 0.0f, o1 = 0.0f, o2 = 0.0f;
#pragma unroll
    for (int t = 0; t < 8; ++t) {
        float al = w[t] * inv;
        const float* f = &feat[mi[t] * 3];
        o0 += al * f[0];  o1 += al * f[1];  o2 += al * f[2];
    }

    if (hif == 0) {                       // one writer per pixel, [C,H,W] layout
        out[p]          = o0;
        out[HW + p]     = o1;
        out[2 * HW + p] = o2;
    }
}

extern "C" void kernel_launch(void* const* d_in, const int* in_sizes, int n_in,
                              void* d_out, int out_size, void* d_ws, size_t ws_size,
                              hipStream_t stream) {
    (void)in_sizes; (void)n_in; (void)d_ws; (void)ws_size; (void)out_size;
    const float* xy    = (const float*)d_in[0];
    const float* scale = (const float*)d_in[1];
    const float* rot   = (const float*)d_in[2];
    const float* feat  = (const float*)d_in[3];
    // d_in[4] = topk (==8), baked into the kernel.
    float* out = (float*)d_out;

    // 512 blocks x 256 threads; each block covers 128 consecutive pixels.
    gsplat_wmma_kernel<<<dim3(HW / 128), dim3(256), 0, stream>>>(xy, scale, rot, feat, out);
}